// GS_model_79714593013808
// MI455X (gfx1250) — compile-verified
//
#include <hip/hip_runtime.h>
#include <hip/hip_bf16.h>
#include <math.h>

typedef __attribute__((ext_vector_type(2))) float v2f;
typedef __attribute__((ext_vector_type(8))) float v8f;

// ---------------------------------------------------------------------------
// Utility: zero-fill (graph-capture safe, no hipMemset*)
// ---------------------------------------------------------------------------
__global__ void fill_zero_kernel(float* __restrict__ p, long long n) {
  long long i = (long long)blockIdx.x * blockDim.x + threadIdx.x;
  long long stride = (long long)gridDim.x * blockDim.x;
  for (; i < n; i += stride) p[i] = 0.0f;
}

// ---------------------------------------------------------------------------
// Degree count + reciprocal (mean with 0 for isolated nodes => 1/max(cnt,1))
// ---------------------------------------------------------------------------
__global__ void degree_kernel(const long long* __restrict__ dst,
                              float* __restrict__ cnt, int n_edges) {
  int e = blockIdx.x * blockDim.x + threadIdx.x;
  if (e < n_edges) atomicAdd(&cnt[(int)dst[e]], 1.0f);
}

__global__ void recip_kernel(float* __restrict__ cnt, int n_nodes) {
  int i = blockIdx.x * blockDim.x + threadIdx.x;
  if (i < n_nodes) cnt[i] = 1.0f / fmaxf(cnt[i], 1.0f);
}

// ---------------------------------------------------------------------------
// Edge scatter: agg[dst] += xin[src]   (float4 gather, f32 atomics into L2)
// Consecutive lanes take consecutive 16B chunks of the same edge => coalesced
// gather of a full feature row per 1 (F=128) or 2 (F=256) waves.
// ---------------------------------------------------------------------------
template <int F>
__global__ void scatter_add_kernel(const float* __restrict__ xin,
                                   const long long* __restrict__ src,
                                   const long long* __restrict__ dst,
                                   float* __restrict__ agg, int n_edges) {
  constexpr int CH = F / 4;  // float4 chunks per row
  long long idx = (long long)blockIdx.x * blockDim.x + threadIdx.x;
  int e = (int)(idx / CH);
  int c = (int)(idx % CH);
  if (e >= n_edges) return;
  int s = (int)src[e];
  int d = (int)dst[e];
  const float4 v = *(const float4*)(xin + (size_t)s * F + (size_t)c * 4);
  float* p = agg + (size_t)d * F + (size_t)c * 4;
  atomicAdd(p + 0, v.x);
  atomicAdd(p + 1, v.y);
  atomicAdd(p + 2, v.z);
  atomicAdd(p + 3, v.w);
}

// ---------------------------------------------------------------------------
// Fused SAGE layer GEMM using full-precision f32 WMMA (16x16x4):
//   out[i, o] = sum_k agg[i,k]*invc[i]*Wl[o,k] + sum_k xin[i,k]*Wr[o,k] + b[o]
// One wave -> one 16(M)x16(N) output tile; block = 4 waves -> 64 rows.
//
// Fragment layouts per CDNA5 ISA 7.12.2 (wave32):
//   A (16x4 f32):  lanes 0-15: M=lane,   VGPR0/1 = K=0/1
//                  lanes16-31: M=lane-16,VGPR0/1 = K=2/3
//   B (4x16 f32):  rows striped across lanes within a VGPR:
//                  VGPR0 = rows K=0 (lanes 0-15) / K=2 (lanes 16-31)
//                  VGPR1 = rows K=1 / K=3 ; N = lane&15
//   C/D (16x16):   VGPR v: M = v (lanes 0-15) or v+8 (lanes 16-31); N = lane&15
// ---------------------------------------------------------------------------
template <int FIN, int FOUT, bool RELU>
__global__ __launch_bounds__(128) void sage_gemm_wmma_kernel(
    const float* __restrict__ xin, const float* __restrict__ agg,
    const float* __restrict__ invc, const float* __restrict__ Wl,
    const float* __restrict__ Wr, const float* __restrict__ bias,
    float* __restrict__ out, int n_nodes) {
  const int lane = threadIdx.x;          // 0..31 (wave32)
  const int l15 = lane & 15;
  const int hi = lane >> 4;              // 0 or 1
  const int khalf = hi * 2;              // K sub-offset for A/B fragments

  const int mrow = blockIdx.x * 64 + threadIdx.y * 16 + l15;
  const int mr = mrow < n_nodes ? mrow : (n_nodes - 1);  // clamp for safe loads
  const int ncol = blockIdx.y * 16 + l15;
  const int nc = ncol < FOUT ? ncol : (FOUT - 1);        // clamp for safe loads

  const float ic = invc[mr];
  const float* pa1 = agg + (size_t)mr * FIN + khalf;   // mean path A
  const float* pa2 = xin + (size_t)mr * FIN + khalf;   // root path A
  const float* pb1 = Wl + (size_t)nc * FIN + khalf;    // mean path B (= Wl^T col)
  const float* pb2 = Wr + (size_t)nc * FIN + khalf;    // root path B

  v8f acc = {};

  // K pass 1: mean-aggregated features x Wl^T
#pragma unroll 4
  for (int k = 0; k < FIN; k += 4) {
    v2f a, b;
    a.x = pa1[k] * ic;
    a.y = pa1[k + 1] * ic;
    b.x = pb1[k];
    b.y = pb1[k + 1];
    acc = __builtin_amdgcn_wmma_f32_16x16x4_f32(false, a, false, b, (short)0,
                                                acc, false, false);
  }
  // K pass 2: root features x Wr^T
#pragma unroll 4
  for (int k = 0; k < FIN; k += 4) {
    v2f a, b;
    a.x = pa2[k];
    a.y = pa2[k + 1];
    b.x = pb2[k];
    b.y = pb2[k + 1];
    acc = __builtin_amdgcn_wmma_f32_16x16x4_f32(false, a, false, b, (short)0,
                                                acc, false, false);
  }

  // Epilogue: bias (+ ReLU), guarded stores.
  if (ncol < FOUT) {
    const float bv = bias[ncol];
    const int rowbase = blockIdx.x * 64 + threadIdx.y * 16 + hi * 8;
#pragma unroll
    for (int v = 0; v < 8; ++v) {
      const int m = rowbase + v;
      if (m < n_nodes) {
        float r = acc[v] + bv;
        if (RELU) r = fmaxf(r, 0.0f);
        out[(size_t)m * FOUT + ncol] = r;
      }
    }
  }
}

// ---------------------------------------------------------------------------
// Row-wise log_softmax over 10 logits, in place on d_out.
// ---------------------------------------------------------------------------
__global__ void log_softmax10_kernel(float* __restrict__ out, int n_nodes) {
  int i = blockIdx.x * blockDim.x + threadIdx.x;
  if (i >= n_nodes) return;
  float* r = out + (size_t)i * 10;
  float v[10];
  float m = -INFINITY;
#pragma unroll
  for (int j = 0; j < 10; ++j) {
    v[j] = r[j];
    m = fmaxf(m, v[j]);
  }
  float s = 0.0f;
#pragma unroll
  for (int j = 0; j < 10; ++j) s += __expf(v[j] - m);
  const float ls = __logf(s);
#pragma unroll
  for (int j = 0; j < 10; ++j) r[j] = v[j] - m - ls;
}

// ---------------------------------------------------------------------------
// Host-side launch sequence (graph-capture safe: kernels only, all on stream)
// ---------------------------------------------------------------------------
extern "C" void kernel_launch(void* const* d_in, const int* in_sizes, int n_in,
                              void* d_out, int out_size, void* d_ws,
                              size_t ws_size, hipStream_t stream) {
  (void)n_in; (void)out_size; (void)ws_size;

  constexpr int IN_F = 128, HID = 256, OUTF = 10;

  const float* x = (const float*)d_in[0];
  const long long* edge = (const long long*)d_in[1];  // int64 in reference
  const float* W1l = (const float*)d_in[2];
  const float* b1l = (const float*)d_in[3];
  const float* W1r = (const float*)d_in[4];
  const float* W2l = (const float*)d_in[5];
  const float* b2l = (const float*)d_in[6];
  const float* W2r = (const float*)d_in[7];
  const float* W3l = (const float*)d_in[8];
  const float* b3l = (const float*)d_in[9];
  const float* W3r = (const float*)d_in[10];

  const int N = in_sizes[0] / IN_F;       // 50000
  const int E = in_sizes[1] / 2;          // 800000
  const long long* src = edge;            // edge_index[0]
  const long long* dst = edge + E;        // edge_index[1]

  // Workspace layout (floats): invc[N] | agg[N*HID] | h1[N*HID] | h2[N*HID]
  float* ws = (float*)d_ws;
  const long long Npad = ((long long)N + 255) / 256 * 256;
  float* invc = ws;
  float* agg = ws + Npad;
  float* h1 = agg + (long long)N * HID;
  float* h2 = h1 + (long long)N * HID;

  float* outp = (float*)d_out;

  const int TB = 256;
  const int fillGrid = 2048;

  // --- degrees -> 1/max(deg,1) (recomputed every call: deterministic) ---
  fill_zero_kernel<<<fillGrid, TB, 0, stream>>>(invc, (long long)N);
  degree_kernel<<<(E + TB - 1) / TB, TB, 0, stream>>>(dst, invc, E);
  recip_kernel<<<(N + TB - 1) / TB, TB, 0, stream>>>(invc, N);

  const dim3 gblock(32, 4);

  // --- Layer 1: [N,128] -> [N,256], ReLU ---
  fill_zero_kernel<<<fillGrid, TB, 0, stream>>>(agg, (long long)N * IN_F);
  {
    long long work = (long long)E * (IN_F / 4);
    scatter_add_kernel<IN_F><<<(unsigned)((work + TB - 1) / TB), TB, 0,
                               stream>>>(x, src, dst, agg, E);
  }
  {
    dim3 grid((N + 63) / 64, HID / 16);
    sage_gemm_wmma_kernel<IN_F, HID, true><<<grid, gblock, 0, stream>>>(
        x, agg, invc, W1l, W1r, b1l, h1, N);
  }

  // --- Layer 2: [N,256] -> [N,256], ReLU ---
  fill_zero_kernel<<<fillGrid, TB, 0, stream>>>(agg, (long long)N * HID);
  {
    long long work = (long long)E * (HID / 4);
    scatter_add_kernel<HID><<<(unsigned)((work + TB - 1) / TB), TB, 0,
                              stream>>>(h1, src, dst, agg, E);
  }
  {
    dim3 grid((N + 63) / 64, HID / 16);
    sage_gemm_wmma_kernel<HID, HID, true><<<grid, gblock, 0, stream>>>(
        h1, agg, invc, W2l, W2r, b2l, h2, N);
  }

  // --- Layer 3: [N,256] -> [N,10], then log_softmax ---
  fill_zero_kernel<<<fillGrid, TB, 0, stream>>>(agg, (long long)N * HID);
  {
    long long work = (long long)E * (HID / 4);
    scatter_add_kernel<HID><<<(unsigned)((work + TB - 1) / TB), TB, 0,
                              stream>>>(h2, src, dst, agg, E);
  }
  {
    dim3 grid((N + 63) / 64, (OUTF + 15) / 16);
    sage_gemm_wmma_kernel<HID, OUTF, false><<<grid, gblock, 0, stream>>>(
        h2, agg, invc, W3l, W3r, b3l, outp, N);
  }
  log_softmax10_kernel<<<(N + TB - 1) / TB, TB, 0, stream>>>(outp, N);
}